// FLGC_84344567758934
// MI455X (gfx1250) — compile-verified
//
#include <hip/hip_runtime.h>

typedef __attribute__((ext_vector_type(2))) float v2f;
typedef __attribute__((ext_vector_type(8))) float v8f;

#define B_     16
#define CIN_   64
#define COUT_  64
#define H_     224
#define WSP_   224
#define G_     8
#define HW_    (H_ * WSP_)       // 50176 spatial positions per (b, channel)
#define NTILE_ (HW_ / 16)        // 3136 column tiles of 16 per image
#define WAVES_PER_BLOCK_ 8       // 256 threads, wave32

// ---------------------------------------------------------------------------
// Kernel 1: fold the soft group-routing into the 1x1 weights.
//   W_eff[o,c] = W[o,c] * (softmax_g T[o,:]) . (softmax_g S[c,:])
// 64 blocks (o) x 64 threads (c); trivial cost, result lives in d_ws (16 KB).
// ---------------------------------------------------------------------------
__global__ __launch_bounds__(64) void flgc_build_weff(
    const float* __restrict__ W, const float* __restrict__ S,
    const float* __restrict__ T, float* __restrict__ Weff) {
  const int o = blockIdx.x;
  const int c = threadIdx.x;

  float tv[G_], sv[G_];
  float tmax = -3.4e38f, smax = -3.4e38f;
#pragma unroll
  for (int g = 0; g < G_; ++g) { tv[g] = T[o * G_ + g]; tmax = fmaxf(tmax, tv[g]); }
#pragma unroll
  for (int g = 0; g < G_; ++g) { sv[g] = S[c * G_ + g]; smax = fmaxf(smax, sv[g]); }
  float tsum = 0.f, ssum = 0.f;
#pragma unroll
  for (int g = 0; g < G_; ++g) { tv[g] = __expf(tv[g] - tmax); tsum += tv[g]; }
#pragma unroll
  for (int g = 0; g < G_; ++g) { sv[g] = __expf(sv[g] - smax); ssum += sv[g]; }
  float mix = 0.f;
  const float inv = 1.0f / (tsum * ssum);
#pragma unroll
  for (int g = 0; g < G_; ++g) mix += tv[g] * sv[g];
  mix *= inv;

  Weff[o * CIN_ + c] = W[o * CIN_ + c] * mix;
}

// ---------------------------------------------------------------------------
// Kernel 2: D[64, N] = W_eff[64,64] x X[64, N], N = B*H*W.
// One wave -> one 16-wide spatial column tile, all 64 output rows.
// K=64 reduction = 16 x V_WMMA_F32_16X16X4_F32 per 16x16 tile, 4 M-tiles.
// x is streamed once (NT loads); out streamed once (NT stores).
// ---------------------------------------------------------------------------
__global__ __launch_bounds__(256) void flgc_gemm_wmma(
    const float* __restrict__ x, const float* __restrict__ Weff,
    float* __restrict__ out) {
  const int lane   = threadIdx.x & 31;
  const int waveId = blockIdx.x * WAVES_PER_BLOCK_ + (threadIdx.x >> 5);

  const int b   = waveId / NTILE_;          // image
  const int nt  = waveId % NTILE_;          // 16-column tile within image
  const int nlo = lane & 15;                // N index within tile
  const int half = lane >> 4;               // 0: K pair {0,1}; 1: K pair {2,3}

  // B-operand base: x[b, half*2, nt*16 + nlo]; k-step adds channel 4k (+1).
  const float* xp = x + (long long)b * CIN_ * HW_
                      + (long long)(half * 2) * HW_
                      + (long long)nt * 16 + nlo;

  // A-operand base: W_eff[row = nlo (+16m), col = half*2 (+4k)], float2 pair.
  const float* wp = Weff + nlo * CIN_ + half * 2;

  v8f acc0 = {}, acc1 = {}, acc2 = {}, acc3 = {};

#pragma unroll
  for (int k = 0; k < 16; ++k) {
    // B tile (4 x 16): two channels per lane-half, single-use data -> NT loads.
    v2f bm;
    bm.x = __builtin_nontemporal_load(xp + (long long)(4 * k)     * HW_);
    bm.y = __builtin_nontemporal_load(xp + (long long)(4 * k + 1) * HW_);

    // A tiles (16 x 4) for the four M blocks; 16 KB table, stays cached.
    v2f a0 = *(const v2f*)(wp + 0 * 16 * CIN_ + 4 * k);
    v2f a1 = *(const v2f*)(wp + 1 * 16 * CIN_ + 4 * k);
    v2f a2 = *(const v2f*)(wp + 2 * 16 * CIN_ + 4 * k);
    v2f a3 = *(const v2f*)(wp + 3 * 16 * CIN_ + 4 * k);

    acc0 = __builtin_amdgcn_wmma_f32_16x16x4_f32(false, a0, false, bm,
                                                 (short)0, acc0, false, false);
    acc1 = __builtin_amdgcn_wmma_f32_16x16x4_f32(false, a1, false, bm,
                                                 (short)0, acc1, false, false);
    acc2 = __builtin_amdgcn_wmma_f32_16x16x4_f32(false, a2, false, bm,
                                                 (short)0, acc2, false, false);
    acc3 = __builtin_amdgcn_wmma_f32_16x16x4_f32(false, a3, false, bm,
                                                 (short)0, acc3, false, false);
  }

  // C/D layout: VGPR j holds rows {j, j+8} (lane halves), cols = nlo.
  float* op = out + (long long)b * COUT_ * HW_ + (long long)nt * 16 + nlo;
#pragma unroll
  for (int j = 0; j < 8; ++j) {
    const int row = j + half * 8;
    __builtin_nontemporal_store(acc0[j], op + (long long)(row +  0) * HW_);
    __builtin_nontemporal_store(acc1[j], op + (long long)(row + 16) * HW_);
    __builtin_nontemporal_store(acc2[j], op + (long long)(row + 32) * HW_);
    __builtin_nontemporal_store(acc3[j], op + (long long)(row + 48) * HW_);
  }
}

extern "C" void kernel_launch(void* const* d_in, const int* in_sizes, int n_in,
                              void* d_out, int out_size, void* d_ws, size_t ws_size,
                              hipStream_t stream) {
  const float* x = (const float*)d_in[0];   // [16, 64, 224, 224]
  const float* W = (const float*)d_in[1];   // [64, 64, 1, 1]
  const float* S = (const float*)d_in[2];   // [64, 8]
  const float* T = (const float*)d_in[3];   // [64, 8]
  float* out  = (float*)d_out;              // [16, 64, 224, 224]
  float* Weff = (float*)d_ws;               // 64*64 floats = 16 KB scratch

  flgc_build_weff<<<dim3(COUT_), dim3(CIN_), 0, stream>>>(W, S, T, Weff);

  const int total_waves = B_ * NTILE_;                       // 50176
  const int blocks = total_waves / WAVES_PER_BLOCK_;         // 6272
  flgc_gemm_wmma<<<dim3(blocks), dim3(256), 0, stream>>>(x, Weff, out);
}